// BahdanauAttention_30236569764263
// MI455X (gfx1250) — compile-verified
//
#include <hip/hip_runtime.h>
#include <math.h>

typedef __attribute__((ext_vector_type(2))) float v2f;
typedef __attribute__((ext_vector_type(8))) float v8f;

#define B_  16
#define TQ  128
#define TK  256
#define DIN 64
#define H_  256
#define DV  256
#define NEGV (-1000000.0f)

__device__ __forceinline__ float fast_tanhf(float x) {
#if __has_builtin(__builtin_amdgcn_tanhf)
  return __builtin_amdgcn_tanhf(x);     // v_tanh_f32 (gfx1250 TRANS op)
#else
  return tanhf(x);
#endif
}

__device__ __forceinline__ v8f wmma_f32_16x16x4(v2f a, v2f b, v8f c) {
  // (neg_a, A, neg_b, B, c_mod, C, reuse_a, reuse_b)
  return __builtin_amdgcn_wmma_f32_16x16x4_f32(false, a, false, b, (short)0, c,
                                               false, false);
}

// ---------------------------------------------------------------------------
// Y[rows,H_] = X[rows,DIN] @ W[DIN,H_]; one 16x16 D-tile per wave, WMMA f32.
// ---------------------------------------------------------------------------
__global__ __launch_bounds__(256) void proj_kernel(const float* __restrict__ X,
                                                   const float* __restrict__ W,
                                                   float* __restrict__ Y,
                                                   int rows) {
  const int lane = threadIdx.x & 31;
  const int wave = threadIdx.x >> 5;
  const int tile = blockIdx.x * 8 + wave;
  const int ntn  = H_ / 16;
  const int m0   = (tile / ntn) * 16;
  const int n0   = (tile % ntn) * 16;
  if (m0 >= rows) return;                   // uniform per wave

  const int mrow = lane & 15;
  const int koff = (lane >> 4) * 2;
  v8f acc = {};
  for (int k0 = 0; k0 < DIN; k0 += 4) {
    v2f a, b;
    a.x = X[(m0 + mrow) * DIN + k0 + koff];
    a.y = X[(m0 + mrow) * DIN + k0 + koff + 1];
    b.x = W[(k0 + koff) * H_ + n0 + mrow];
    b.y = W[(k0 + koff + 1) * H_ + n0 + mrow];
    acc = wmma_f32_16x16x4(a, b, acc);
  }
  const int mbase = (lane >> 4) * 8;
  for (int j = 0; j < 8; ++j)
    Y[(m0 + mbase + j) * H_ + n0 + mrow] = acc[j];
}

// ---------------------------------------------------------------------------
// Score stage (the tanh wall): grid (TQ/16, TK/64, B_) = 512 WGs so the whole
// chip is covered ~2x deep. Each WG: 16 q-rows x 64 k-cols, full H reduction.
// scores[b,q,k] = sum_h wv[h]*tanh(qp[q,h]+kp[k,h])
// ---------------------------------------------------------------------------
__global__ __launch_bounds__(256) void score_kernel(
    const float* __restrict__ qp,        // [B_,TQ,H_]
    const float* __restrict__ kp,        // [B_,TK,H_]
    const float* __restrict__ wv,        // [H_]
    float*       __restrict__ scores)    // [B_,TQ,TK]
{
  __shared__ float qp_lds[H_ * 17];      // [h][q], padded
  __shared__ float kp_lds[H_ * 65];      // [h][k], padded
  __shared__ float wv_lds[H_];

  const int t     = threadIdx.x;
  const int qbase = blockIdx.x * 16;
  const int kbase = blockIdx.y * 64;
  const int b     = blockIdx.z;

  wv_lds[t] = wv[t];
  for (int i = t; i < 16 * H_; i += 256) {            // coalesced over h
    int q = i >> 8, h = i & 255;
    qp_lds[h * 17 + q] = qp[(b * TQ + qbase + q) * H_ + h];
  }
  for (int i = t; i < 64 * H_; i += 256) {            // coalesced over h
    int k = i >> 8, h = i & 255;
    kp_lds[h * 65 + k] = kp[(b * TK + kbase + k) * H_ + h];
  }
  __syncthreads();

  const int q  = t & 15;
  const int kg = t >> 4;                               // owns 4 consecutive k
  float a0 = 0.f, a1 = 0.f, a2 = 0.f, a3 = 0.f;
  const float* kq = &kp_lds[kg * 4];
  for (int h = 0; h < H_; ++h) {
    float qv = qp_lds[h * 17 + q];                     // LDS broadcast
    float w  = wv_lds[h];                              // LDS broadcast
    a0 += w * fast_tanhf(qv + kq[h * 65 + 0]);
    a1 += w * fast_tanhf(qv + kq[h * 65 + 1]);
    a2 += w * fast_tanhf(qv + kq[h * 65 + 2]);
    a3 += w * fast_tanhf(qv + kq[h * 65 + 3]);
  }
  float4 r = make_float4(a0, a1, a2, a3);
  *(float4*)&scores[(size_t)(b * TQ + qbase + q) * TK + kbase + kg * 4] = r;  // b128
}

// ---------------------------------------------------------------------------
// Masked softmax (parallel, 16 thr/row) + attn@V via WMMA f32 16x16x4.
// ---------------------------------------------------------------------------
__global__ __launch_bounds__(256) void softmax_av_kernel(
    const float* __restrict__ scores,    // [B_,TQ,TK]
    const float* __restrict__ V,         // [B_,TK,DV]
    const int*   __restrict__ valid_lens,
    float*       __restrict__ out)       // [B_,TQ,DV]
{
  __shared__ float sc_lds[16 * 257];     // [q][k], padded
  __shared__ float red[16 * 17];
  __shared__ float rowstat[16];

  const int t     = threadIdx.x;
  const int qbase = blockIdx.x * 16;
  const int b     = blockIdx.y;
  const int valid = valid_lens[b];

  for (int i = t; i < 16 * TK; i += 256) {             // coalesced, mask on load
    int q = i >> 8, k = i & 255;
    float s = scores[(size_t)(b * TQ + qbase + q) * TK + k];
    sc_lds[q * 257 + k] = (k < valid) ? s : NEGV;
  }
  __syncthreads();

  const int q16 = t >> 4, l = t & 15;                  // 16 threads per row
  float pm = NEGV;
  for (int j = 0; j < 16; ++j)
    pm = fmaxf(pm, sc_lds[q16 * 257 + l + j * 16]);
  red[q16 * 17 + l] = pm;
  __syncthreads();
  if (t < 16) {
    float m = red[t * 17];
    for (int i = 1; i < 16; ++i) m = fmaxf(m, red[t * 17 + i]);
    rowstat[t] = m;
  }
  __syncthreads();

  const float m = rowstat[q16];
  float ps = 0.f;
  for (int j = 0; j < 16; ++j) {
    int idx = q16 * 257 + l + j * 16;
    float e = __expf(sc_lds[idx] - m);                 // v_exp_f32
    sc_lds[idx] = e;
    ps += e;
  }
  red[q16 * 17 + l] = ps;
  __syncthreads();
  if (t < 16) {
    float s = 0.f;
    for (int i = 0; i < 16; ++i) s += red[t * 17 + i];
    rowstat[t] = 1.0f / s;
  }
  __syncthreads();
  const float inv = rowstat[q16];
  for (int j = 0; j < 16; ++j)
    sc_lds[q16 * 257 + l + j * 16] *= inv;
  __syncthreads();

  // out[16,DV] = attn[16,TK] @ V[b]; 8 waves x 2 N-tiles, K=256 in steps of 4
  const int lane  = t & 31;
  const int wave  = t >> 5;
  const int mrow  = lane & 15;
  const int koff  = (lane >> 4) * 2;
  const int mbase = (lane >> 4) * 8;
  for (int nt = wave * 2; nt < wave * 2 + 2; ++nt) {
    const int n0 = nt * 16;
    v8f acc = {};
    for (int k0 = 0; k0 < TK; k0 += 4) {
      v2f a, bv;
      a.x  = sc_lds[mrow * 257 + k0 + koff];
      a.y  = sc_lds[mrow * 257 + k0 + koff + 1];
      bv.x = V[(size_t)(b * TK + k0 + koff) * DV + n0 + mrow];
      bv.y = V[(size_t)(b * TK + k0 + koff + 1) * DV + n0 + mrow];
      acc = wmma_f32_16x16x4(a, bv, acc);
    }
    for (int j = 0; j < 8; ++j)
      out[(size_t)(b * TQ + qbase + mbase + j) * DV + n0 + mrow] = acc[j];
  }
}

extern "C" void kernel_launch(void* const* d_in, const int* in_sizes, int n_in,
                              void* d_out, int out_size, void* d_ws, size_t ws_size,
                              hipStream_t stream) {
  const float* queries    = (const float*)d_in[0];
  const float* keys       = (const float*)d_in[1];
  const float* values     = (const float*)d_in[2];
  const int*   valid_lens = (const int*)d_in[3];
  const float* wq         = (const float*)d_in[4];
  const float* wk         = (const float*)d_in[5];
  const float* wv         = (const float*)d_in[6];
  float* out = (float*)d_out;

  float* qp     = (float*)d_ws;                        // 2 MB
  float* kp     = qp + (size_t)B_ * TQ * H_;           // 4 MB
  float* scores = kp + (size_t)B_ * TK * H_;           // 2 MB

  proj_kernel<<<(B_ * TQ / 16) * (H_ / 16) / 8, 256, 0, stream>>>(queries, wq, qp, B_ * TQ);
  proj_kernel<<<(B_ * TK / 16) * (H_ / 16) / 8, 256, 0, stream>>>(keys,    wk, kp, B_ * TK);

  score_kernel<<<dim3(TQ / 16, TK / 64, B_), 256, 0, stream>>>(qp, kp, wv, scores);
  softmax_av_kernel<<<dim3(TQ / 16, B_), 256, 0, stream>>>(scores, values, valid_lens, out);
}